// QuantumConv2d_76321568850110
// MI455X (gfx1250) — compile-verified
//
#include <hip/hip_runtime.h>
#include <hip/hip_bf16.h>

typedef __attribute__((ext_vector_type(2))) float v2f;
typedef __attribute__((ext_vector_type(8))) float v8f;
typedef __attribute__((ext_vector_type(4))) unsigned int v4u;
typedef __attribute__((ext_vector_type(8))) int v8i;
typedef __attribute__((ext_vector_type(4))) int v4i;

#define KH_ 3
#define OC_ 32
#define IC_ 16
#define H_  128
#define W_  128
#define NB_ 8

#define XLINE_ 144   // 128 data dwords + 16 pad dwords (TDM pad: bank-decorrelates j-planes)

// out[n,i,h,w] = (1/3) sum_{j,k} cos(R) cos(th) - sin(R) sin(th)
// GEMM: M=32 (i), K=96, N=pixels, via V_WMMA_F32_16X16X4_F32.
// K ordering, chunk of 12 per j-pair jj (j0=2jj, j1=2jj+1), step s=k:
//   p = jj*12 + 4k + q : q=0: cos(j0,k)  q=1: sin(j0,k)  q=2: cos(j1,k)  q=3: sin(j1,k)
// => B-fragment rows K0/K1 (low lanes) use only j0, rows K2/K3 (high lanes) only j1:
//    every lane evaluates sincos for exactly one j per pair. Weights match:
//    q even -> cos(th)/3 ; q odd -> -sin(th)/3.

__global__ __launch_bounds__(256) void qconv_wmma_tdm_kernel(
    const float* __restrict__ x,      // (8,16,128,128)
    const float* __restrict__ theta,  // (1536,3)
    float* __restrict__ out)          // (8,32,128,128)
{
    __shared__ float lds_w[96 * 32];        // weight matrix [p][i], 12 KB
    __shared__ float lds_x[48 * XLINE_];    // staged x: line = k*16+j, 27.6 KB

    const int tid = threadIdx.x;
    const int nh  = blockIdx.x;             // 1024 blocks = 8 n * 128 h
    const int n   = nh >> 7;
    const int h   = nh & 127;

    const int h0   = (h > 0) ? (h - 1) : 0;
    const int rows = (h > 0 && h < 127) ? 3 : 2;

    // ---- TDM: DMA the 16(j) x rows(h) x 128(w) fp32 tile into LDS ----
    if (tid == 0) {
        const unsigned lds_off =
            (unsigned)(uintptr_t)(void*)lds_x + (h == 0 ? 16u * XLINE_ * 4u : 0u);
        const unsigned long long gaddr =
            (unsigned long long)(uintptr_t)x +
            ((unsigned long long)(n * IC_ * H_ * W_ + h0 * W_) << 2);

        // D# group 0: count=1, lds_addr, global_addr[56:0], type=2
        v4u g0 = { 1u, lds_off, (unsigned)(gaddr & 0xFFFFFFFFu),
                   (unsigned)(((gaddr >> 32) & 0x1FFFFFFu) | (2u << 30)) };
        // D# group 1: data_size=4B, pad_enable, pad_interval=128dw, pad_amount=16dw,
        //   tensor_dim0=128, tensor_dim1=16, tile_dim0=128, tile_dim1=16, tile_dim2=rows,
        //   dim0_stride=16384 (j planes), dim1_stride=128 (h rows)
        v8i g1 = { (int)0x1F920000,
                   (int)(128u << 16),
                   (int)(16u  << 16),
                   (int)(128u << 16),
                   (int)(16u | ((unsigned)rows << 16)),
                   (int)16384,
                   (int)(128u << 16),
                   0 };
        v4i g2 = { (int)(128 - h0), 0, 0, 0 };   // tensor_dim2 (>= rows)
        v4i g3 = { 0, 0, 0, 0 };
#if defined(__clang_major__) && (__clang_major__ >= 23)
        v8i g1b = { 0, 0, 0, 0, 0, 0, 0, 0 };
        __builtin_amdgcn_tensor_load_to_lds(g0, g1, g2, g3, g1b, 0);
#else
        __builtin_amdgcn_tensor_load_to_lds(g0, g1, g2, g3, 0);
#endif
        __builtin_amdgcn_s_wait_tensorcnt(0);
    }

    // ---- zero-fill missing boundary k-slot (disjoint from TDM target) ----
    if (h == 0) {
        for (int idx = tid; idx < 16 * 128; idx += 256)
            lds_x[(idx >> 7) * XLINE_ + (idx & 127)] = 0.0f;          // slot k=0
    } else if (h == 127) {
        for (int idx = tid; idx < 16 * 128; idx += 256)
            lds_x[(32 + (idx >> 7)) * XLINE_ + (idx & 127)] = 0.0f;   // slot k=2
    }

    // ---- build weight matrix in LDS ----
    for (int idx = tid; idx < 96 * 32; idx += 256) {
        const int i  = idx & 31;
        const int p  = idx >> 5;
        const int jj = p / 12;
        const int r  = p % 12;
        const int k  = r >> 2;
        const int q  = r & 3;
        const int j  = 2 * jj + (q >> 1);
        const float th = theta[(i * j) * 3 + k];
        float sv, cv;
        __sincosf(th, &sv, &cv);
        lds_w[p * 32 + i] = ((q & 1) == 0) ? (cv * (1.0f / 3.0f))
                                           : (-sv * (1.0f / 3.0f));
    }

    asm volatile("" ::: "memory");
    __syncthreads();

    // ---- per-wave: 16 pixels x 32 OC ----
    const int wave = tid >> 5;
    const int lane = tid & 31;
    const int t    = lane & 15;
    const int hi01 = (lane >= 16) ? 1 : 0;
    const int hsel = hi01 * 2;              // K-subrow select (B rows K0/K2 split)

    const int w0 = wave * 16;
    const int w  = w0 + t;

    v8f c0 = {};  // OC 0..15
    v8f c1 = {};  // OC 16..31

#pragma unroll
    for (int jj = 0; jj < 8; ++jj) {
        const int jm = 2 * jj + hi01;       // this lane's j
        float cs[3], sn[3];
#pragma unroll
        for (int k = 0; k < 3; ++k) {
            const float* __restrict__ lp = &lds_x[(k * 16 + jm) * XLINE_];
            const float a = (w > 0)        ? lp[w - 1] : 0.0f;
            const float b = lp[w];
            const float c = (w < W_ - 1)   ? lp[w + 1] : 0.0f;
            __sincosf(a + b + c, &sn[k], &cs[k]);
        }
#pragma unroll
        for (int s = 0; s < 3; ++s) {
            const int pbase = jj * 12 + s * 4;
            v2f b;
            b.x = cs[s];                    // K row 4s+hsel   : cos(R(jm,s))
            b.y = sn[s];                    // K row 4s+hsel+1 : sin(R(jm,s))
            v2f a0, a1;
            a0.x = lds_w[(pbase + hsel)     * 32 + t];
            a0.y = lds_w[(pbase + hsel + 1) * 32 + t];
            a1.x = lds_w[(pbase + hsel)     * 32 + (t + 16)];
            a1.y = lds_w[(pbase + hsel + 1) * 32 + (t + 16)];
            c0 = __builtin_amdgcn_wmma_f32_16x16x4_f32(
                false, a0, false, b, (short)0, c0, false, false);
            c1 = __builtin_amdgcn_wmma_f32_16x16x4_f32(
                false, a1, false, b, (short)0, c1, false, false);
        }
    }

    // ---- store (write-once -> non-temporal) ----
    float* __restrict__ on = out + (size_t)n * OC_ * H_ * W_;
    const int rowoff = h * W_ + w0 + t;
#pragma unroll
    for (int r = 0; r < 8; ++r) {
        const int i0 = r + hi01 * 8;
        __builtin_nontemporal_store(c0[r], &on[(i0)      * (H_ * W_) + rowoff]);
        __builtin_nontemporal_store(c1[r], &on[(i0 + 16) * (H_ * W_) + rowoff]);
    }
}

extern "C" void kernel_launch(void* const* d_in, const int* in_sizes, int n_in,
                              void* d_out, int out_size, void* d_ws, size_t ws_size,
                              hipStream_t stream) {
    (void)in_sizes; (void)n_in; (void)d_ws; (void)ws_size; (void)out_size;
    const float* x     = (const float*)d_in[0];
    const float* theta = (const float*)d_in[1];
    float* out         = (float*)d_out;

    dim3 grid(NB_ * H_);   // one (n, h) output row per block
    dim3 block(256);       // 8 waves; each wave covers 16 of W=128
    hipLaunchKernelGGL(qconv_wmma_tdm_kernel, grid, block, 0, stream, x, theta, out);
}